// Net_80023830659738
// MI455X (gfx1250) — compile-verified
//
#include <hip/hip_runtime.h>
#include <hip/hip_bf16.h>

typedef float v2f __attribute__((ext_vector_type(2)));
typedef float v8f __attribute__((ext_vector_type(8)));

// ---------------------------------------------------------------------------
// Degree / normalization kernels
// ---------------------------------------------------------------------------
__global__ void __launch_bounds__(256)
gcn_deg_init(float* __restrict__ deg, long long n) {
    long long i = (long long)blockIdx.x * blockDim.x + threadIdx.x;
    if (i < n) deg[i] = 1.0f;   // self-loop contributes 1 to every node's degree
}

__global__ void __launch_bounds__(256)
gcn_deg_count(const long long* __restrict__ dst, float* __restrict__ deg, long long e) {
    long long i = (long long)blockIdx.x * blockDim.x + threadIdx.x;
    if (i < e) atomicAdd(&deg[dst[i]], 1.0f);
}

__global__ void __launch_bounds__(256)
gcn_deg_rsqrt(float* __restrict__ deg, long long n) {
    long long i = (long long)blockIdx.x * blockDim.x + threadIdx.x;
    if (i < n) deg[i] = rsqrtf(deg[i]);   // deg >= 1 always (self loop)
}

// ---------------------------------------------------------------------------
// accum[n,f] = dinv[n]^2 * feat[n,f] (+ bias[f])   — folds in the self-loop
// ---------------------------------------------------------------------------
__global__ void __launch_bounds__(256)
gcn_self_init(const float* __restrict__ feat, const float* __restrict__ dinv,
              const float* __restrict__ bias, float* __restrict__ accum,
              long long n, int F) {
    long long i = (long long)blockIdx.x * blockDim.x + threadIdx.x;
    long long total = n * (long long)F;
    if (i >= total) return;
    long long node = i / F;
    int f = (int)(i - node * F);
    float w = dinv[node];
    w = w * w;
    float v = feat[i] * w;
    if (bias) v += bias[f];
    accum[i] = v;
}

// ---------------------------------------------------------------------------
// Edge scatter: accum[dst] += dinv[src]*dinv[dst] * feat[src]
// One wave per edge; 32 lanes x float4 = 128 features per pass.
// ---------------------------------------------------------------------------
__global__ void __launch_bounds__(256)
gcn_scatter(const float* __restrict__ feat, const long long* __restrict__ src,
            const long long* __restrict__ dst, const float* __restrict__ dinv,
            float* __restrict__ accum, long long e, int F) {
    const int lane = threadIdx.x & 31;
    const int wave = threadIdx.x >> 5;
    long long edge = (long long)blockIdx.x * (blockDim.x >> 5) + wave;
    if (edge >= e) return;                    // wave-uniform
    long long s = src[edge];
    long long d = dst[edge];
    float w = dinv[s] * dinv[d];
    const float* fp = feat + s * (long long)F;
    float* op = accum + d * (long long)F;
    for (int f0 = lane * 4; f0 < F; f0 += 128) {
        float4 v = *(const float4*)(fp + f0);
        atomicAdd(op + f0 + 0, v.x * w);
        atomicAdd(op + f0 + 1, v.y * w);
        atomicAdd(op + f0 + 2, v.z * w);
        atomicAdd(op + f0 + 3, v.w * w);
    }
}

// ---------------------------------------------------------------------------
// GEMM: C[nrows,fout] = act(A[nrows,K] @ W[K,fout] + bias)
// One wave computes one 16x16 tile via V_WMMA_F32_16X16X4_F32 (fp32-exact).
// A-frag (16x4): lane<16 -> row=lane, K={k,k+1}; lane>=16 -> row=lane-16, K={k+2,k+3}
// B-frag (4x16): lane<16 -> col=lane, K={k,k+1}; lane>=16 -> col=lane-16, K={k+2,k+3}
// D (16x16):     VGPR r: lanes0-15 => (M=r, N=lane); lanes16-31 => (M=r+8, N=lane-16)
// ---------------------------------------------------------------------------
__global__ void __launch_bounds__(128)
gcn_gemm_wmma(const float* __restrict__ A, const float* __restrict__ W,
              const float* __restrict__ bias, float* __restrict__ C,
              int nrows, int K, int fout, int relu) {
    const int lane = threadIdx.x & 31;
    const int wave = threadIdx.x >> 5;
    const int tilesN = fout >> 4;
    long long tile = (long long)blockIdx.x * (blockDim.x >> 5) + wave;
    long long totalTiles = (long long)(nrows >> 4) * tilesN;
    if (tile >= totalTiles) return;           // wave-uniform -> EXEC all-1s below

    const int tm = (int)(tile / tilesN);
    const int tn = (int)(tile % tilesN);
    const int half = lane >> 4;               // 0 or 1
    const int l    = lane & 15;

    const float* aptr = A + (long long)(tm * 16 + l) * K + 2 * half;
    const float* bptr = W + (long long)(2 * half) * fout + (tn * 16 + l);

    v8f c = {};
    for (int k = 0; k < K; k += 4) {
        v2f a, b;
        a.x = aptr[k];
        a.y = aptr[k + 1];
        b.x = bptr[(long long)k * fout];
        b.y = bptr[(long long)(k + 1) * fout];
        c = __builtin_amdgcn_wmma_f32_16x16x4_f32(
                /*neg_a=*/false, a, /*neg_b=*/false, b,
                /*c_mod=*/(short)0, c, /*reuse_a=*/false, /*reuse_b=*/false);
    }

    const int colOut = tn * 16 + l;
    float bv = bias ? bias[colOut] : 0.0f;
#pragma unroll
    for (int r = 0; r < 8; ++r) {
        int rowOut = tm * 16 + r + 8 * half;
        float v = c[r] + bv;
        if (relu) v = fmaxf(v, 0.0f);
        C[(long long)rowOut * fout + colOut] = v;
    }
}

// ---------------------------------------------------------------------------
extern "C" void kernel_launch(void* const* d_in, const int* in_sizes, int n_in,
                              void* d_out, int out_size, void* d_ws, size_t ws_size,
                              hipStream_t stream) {
    const float*     x    = (const float*)d_in[0];
    const long long* ei   = (const long long*)d_in[1];
    const float*     W1   = (const float*)d_in[2];
    const float*     b1   = (const float*)d_in[3];
    const float*     W2   = (const float*)d_in[4];
    const float*     b2   = (const float*)d_in[5];
    float*           out  = (float*)d_out;

    const long long HID = in_sizes[3];             // 256
    const long long OUT = in_sizes[5];             // 128
    const long long FIN = in_sizes[2] / HID;       // 128
    const long long N   = in_sizes[0] / FIN;       // 50000
    const long long E   = in_sizes[1] / 2;         // 800000

    const long long* src = ei;
    const long long* dst = ei + E;

    char* ws = (char*)d_ws;
    size_t szN  = ((size_t)(N * 4)       + 255) & ~(size_t)255;
    size_t szNF = ((size_t)(N * FIN * 4) + 255) & ~(size_t)255;
    size_t szNH = ((size_t)(N * HID * 4) + 255) & ~(size_t)255;
    float* dinv = (float*)(ws);
    float* agg1 = (float*)(ws + szN);                // [N, FIN]  = Â X
    float* h1   = (float*)(ws + szN + szNF);         // [N, HID]  = relu(ÂX W1 + b1)
    float* t    = (float*)(ws + szN + szNF + szNH);  // [N, OUT]  = h1 W2
    (void)ws_size; (void)n_in; (void)out_size;

    const int tpb = 256;
    const int wpb = tpb / 32;

    // normalization: deg -> dinv (in place)
    gcn_deg_init <<<(int)((N + tpb - 1) / tpb), tpb, 0, stream>>>(dinv, N);
    gcn_deg_count<<<(int)((E + tpb - 1) / tpb), tpb, 0, stream>>>(dst, dinv, E);
    gcn_deg_rsqrt<<<(int)((N + tpb - 1) / tpb), tpb, 0, stream>>>(dinv, N);

    // layer 1: aggregate X first (128 feats), then GEMM  (ÂX)W1, +b1, relu
    long long totNF = N * FIN;
    gcn_self_init<<<(int)((totNF + tpb - 1) / tpb), tpb, 0, stream>>>(x, dinv, nullptr, agg1, N, (int)FIN);
    gcn_scatter  <<<(int)((E + wpb - 1) / wpb), tpb, 0, stream>>>(x, src, dst, dinv, agg1, E, (int)FIN);

    long long tiles1 = (N / 16) * (HID / 16);
    gcn_gemm_wmma<<<(int)((tiles1 + 3) / 4), 128, 0, stream>>>(agg1, W1, b1, h1, (int)N, (int)FIN, (int)HID, 1);

    // layer 2: GEMM first (h1 W2 -> 128 feats), then aggregate; b2 folded into init
    long long tiles2 = (N / 16) * (OUT / 16);
    gcn_gemm_wmma<<<(int)((tiles2 + 3) / 4), 128, 0, stream>>>(h1, W2, nullptr, t, (int)N, (int)HID, (int)OUT, 0);

    long long totNO = N * OUT;
    gcn_self_init<<<(int)((totNO + tpb - 1) / tpb), tpb, 0, stream>>>(t, dinv, b2, out, N, (int)OUT);
    gcn_scatter  <<<(int)((E + wpb - 1) / wpb), tpb, 0, stream>>>(t, src, dst, dinv, out, E, (int)OUT);
}